// KNN_50199577756191
// MI455X (gfx1250) — compile-verified
//
#include <hip/hip_runtime.h>

// ---- CDNA5 WMMA vector types (f32 16x16x4: A/B = 2 VGPRs, C/D = 8 VGPRs) ----
typedef float v2f __attribute__((ext_vector_type(2)));
typedef float v8f __attribute__((ext_vector_type(8)));

#define KCOLORS 21
#define HW (512 * 512)            // 2^18 pixels per plane
#define IMG_STRIDE (3 * HW)       // floats per image
#define NPIX (16 * HW)            // total pixels
#define TILES_PER_WAVE 8
#define WAVES_PER_BLOCK 2
#define NTHREADS (WAVES_PER_BLOCK * 32)              // 64
#define PX_PER_WAVE (TILES_PER_WAVE * 16)            // 128
#define PX_PER_BLOCK (WAVES_PER_BLOCK * PX_PER_WAVE) // 256
#define KSTRIDE 36  // 144B rows, 16B aligned; halves 32 banks apart; entries 24-31 unused

static __device__ __forceinline__ uint32_t um(uint32_t a, uint32_t b) {
    return a > b ? a : b;         // fuses into v_max3_u32 trees
}

// key = (float_bits & 0xFFFFFFE0) | tag  in ONE VALU op (v_and_or_b32)
static __device__ __forceinline__ uint32_t pack_key(float s, uint32_t tag, uint32_t mask) {
    uint32_t r;
    asm("v_and_or_b32 %0, %1, %2, %3"
        : "=v"(r)
        : "v"(__float_as_uint(s)), "s"(mask), "v"(tag));
    return r;
}

__global__ __launch_bounds__(NTHREADS) void knn_cosine_top1(
    const float* __restrict__ in,      // [16, 3, 512, 512] planar
    const float* __restrict__ colors,  // [21, 3] raw palette
    float* __restrict__ out)           // [16, 3, 512, 512]
{
    __shared__ __align__(16) uint32_t s_k[PX_PER_BLOCK * KSTRIDE]; // 36 KB keys
    __shared__ float s_cn[KCOLORS * 3 + 1];                        // normalized palette

    const int tid  = threadIdx.x;
    const int lane = tid & 31;
    const int wave = tid >> 5;
    const int n    = lane & 15;  // color column (B) / pixel row (A)
    const int hi   = lane >> 4;  // half-wave: K=0,1 vs K=2,3 slots

    // ---- Build B operands in-register: L2-normalized palette, two 16-col tiles
    v2f B0, B1;
    {
        float c0x = colors[3 * n + 0], c0y = colors[3 * n + 1], c0z = colors[3 * n + 2];
        float inv0 = rsqrtf(c0x * c0x + c0y * c0y + c0z * c0z);
        c0x *= inv0; c0y *= inv0; c0z *= inv0;

        float c1x = 0.f, c1y = 0.f, c1z = 0.f;
        if (n + 16 < KCOLORS) {
            c1x = colors[3 * (n + 16) + 0];
            c1y = colors[3 * (n + 16) + 1];
            c1z = colors[3 * (n + 16) + 2];
            float inv1 = rsqrtf(c1x * c1x + c1y * c1y + c1z * c1z);
            c1x *= inv1; c1y *= inv1; c1z *= inv1;
        }
        // lanes 0-15 carry (K0,K1)=(r,g); lanes 16-31 carry (K2,K3)=(b,*).
        // K3 rows are ZERO -> A's K3 slot may hold garbage.
        B0.x = hi ? c0z : c0x;  B0.y = hi ? 0.0f : c0y;
        B1.x = hi ? c1z : c1x;  B1.y = hi ? 0.0f : c1y;

        // Wave 0 caches the normalized palette in LDS for Phase B lookups.
        if (wave == 0 && hi == 0) {
            s_cn[3 * n + 0] = c0x;  s_cn[3 * n + 1] = c0y;  s_cn[3 * n + 2] = c0z;
            if (n + 16 < KCOLORS) {
                s_cn[3 * (n + 16) + 0] = c1x;
                s_cn[3 * (n + 16) + 1] = c1y;
                s_cn[3 * (n + 16) + 2] = c1z;
            }
        }
    }

    const long long blockBase = (long long)blockIdx.x * PX_PER_BLOCK;
    const long long waveBase  = blockBase + (long long)wave * PX_PER_WAVE;
    const int img = (int)(waveBase >> 18);        // / HW (2^18)
    const int off = (int)(waveBase & (HW - 1));   // % HW
    const float* __restrict__ base = in + (long long)img * IMG_STRIDE + off;
    // One half-selected base: lanes 0-15 read the R plane, lanes 16-31 the B plane.
    const float* __restrict__ P0 = base + (hi ? 2 * HW : 0) + n;
    const float* __restrict__ P1 = base + HW + n; // G plane (garbage ok in high half)

    // Per-lane tie-break tags: larger tag == lower color index.
    const uint32_t tag0 = 31u - (uint32_t)n;          // colors 0..15  -> 31..16
    const uint32_t tag1 = 15u - (uint32_t)n;          // colors 16..31 -> 15..0 (24-31 unread)
    const uint32_t mask = 0xFFFFFFE0u;

    // Branch-free LDS bases (entries 0-15 = D0, 16-31 = D1).
    uint32_t* __restrict__ r1base = &s_k[(wave * PX_PER_WAVE + 8 * hi) * KSTRIDE + n];
    uint32_t* __restrict__ r2base = r1base + 16;

    // ---- Phase A: 8 tiles x (2 WMMAs) per wave; packed keys -> LDS ----------
    #pragma unroll
    for (int t = 0; t < TILES_PER_WAVE; ++t) {
        v2f A;
        A.x = P0[t * 16];   // K=0 (r, low half) / K=2 (b, high half)
        A.y = P1[t * 16];   // K=1 (g, low half) / K=3 slot: garbage * 0 in high half

        v8f Z = {};
        // sims[16 px][16 colors]; unnormalized pixel is fine for argmax
        v8f D0 = __builtin_amdgcn_wmma_f32_16x16x4_f32(
            false, A, false, B0, (short)0, Z, false, false);
        v8f D1 = __builtin_amdgcn_wmma_f32_16x16x4_f32(
            false, A, false, B1, (short)0, Z, false, false);

        uint32_t* __restrict__ r1 = r1base + t * 16 * KSTRIDE;
        uint32_t* __restrict__ r2 = r2base + t * 16 * KSTRIDE;
        #pragma unroll
        for (int v = 0; v < 8; ++v) {
            r1[v * KSTRIDE] = pack_key(D0[v], tag0, mask);   // ds_store_2addr pairs
            r2[v * KSTRIDE] = pack_key(D1[v], tag1, mask);   // full wave, no exec branch
        }
    }

    __syncthreads();

    // ---- Phase B: adjacent-pixel pairs per thread ---------------------------
    #pragma unroll
    for (int i = 0; i < 2; ++i) {
        const int pp = 2 * tid + i * (2 * NTHREADS);   // even pixel-pair base
        float ox[2], oy[2], oz[2];

        #pragma unroll
        for (int j = 0; j < 2; ++j) {
            const int p = pp + j;
            const uint32_t* w = &s_k[p * KSTRIDE];
            const uint4 a = *(const uint4*)(w + 0);    // colors  0- 3
            const uint4 b = *(const uint4*)(w + 4);    // colors  4- 7
            const uint4 c = *(const uint4*)(w + 8);    // colors  8-11
            const uint4 d = *(const uint4*)(w + 12);   // colors 12-15
            const uint4 e = *(const uint4*)(w + 16);   // colors 16-19
            const uint4 f = *(const uint4*)(w + 20);   // colors 20-23 (21-23 padded)

            const uint32_t m1 = um(um(um(a.x, a.y), um(a.z, a.w)),
                                   um(um(b.x, b.y), um(b.z, b.w)));
            const uint32_t m2 = um(um(um(c.x, c.y), um(c.z, c.w)),
                                   um(um(d.x, d.y), um(d.z, d.w)));
            const uint32_t m3 = um(um(um(e.x, e.y), um(e.z, e.w)),
                                   um(um(f.x, f.y), um(f.z, f.w)));
            const uint32_t m  = um(um(m1, m2), m3);

            const int  bestk = 31 - (int)(m & 31u);
            const bool zero  = (m >> 5) == 0u;   // all sims == 0 <=> zero pixel

            ox[j] = zero ? 0.f : s_cn[3 * bestk + 0];
            oy[j] = zero ? 0.f : s_cn[3 * bestk + 1];
            oz[j] = zero ? 0.f : s_cn[3 * bestk + 2];
        }

        const long long gp = blockBase + pp;
        const int gimg = (int)(gp >> 18);
        const int go   = (int)(gp & (HW - 1));       // even -> 8B aligned stores
        float* __restrict__ O = out + (long long)gimg * IMG_STRIDE + go;
        float2 vr; vr.x = ox[0]; vr.y = ox[1];
        float2 vg; vg.x = oy[0]; vg.y = oy[1];
        float2 vb; vb.x = oz[0]; vb.y = oz[1];
        *(float2*)(O)          = vr;   // R plane (b64)
        *(float2*)(O + HW)     = vg;   // G plane (b64)
        *(float2*)(O + 2 * HW) = vb;   // B plane (b64)
    }
}

extern "C" void kernel_launch(void* const* d_in, const int* in_sizes, int n_in,
                              void* d_out, int out_size, void* d_ws, size_t ws_size,
                              hipStream_t stream) {
    const float* rgb    = (const float*)d_in[0];  // [16,3,512,512] f32
    const float* colors = (const float*)d_in[1];  // [21,3] f32
    float* outp = (float*)d_out;

    const int grid = NPIX / PX_PER_BLOCK;  // 16384
    knn_cosine_top1<<<grid, NTHREADS, 0, stream>>>(rgb, colors, outp);
    (void)in_sizes; (void)n_in; (void)out_size; (void)d_ws; (void)ws_size;
}